// JKAdvancedGNN_68229850464792
// MI455X (gfx1250) — compile-verified
//
#include <hip/hip_runtime.h>

#define NNODES 50000
#define NEDGES 640000
#define HDIM   128
#define NGRAPH 512
#define LAYERS 2

typedef float v2f __attribute__((ext_vector_type(2)));
typedef float v8f __attribute__((ext_vector_type(8)));

// XOR-swizzled transposed weight layout inside exactly 64 KB of LDS.
// Element W[k][c] lives at ldsW[c*128 + (k ^ ((c&7)<<3) ^ ((c&8)>>1))].
// - (k, k+1) stay adjacent (swizzle never touches bit 0) -> contiguous b64 frag
// - 16 lanes (c = t*16 .. t*16+15) map to 16 distinct bank pairs -> conflict-free
__device__ __forceinline__ int swzW(int c, int k) {
  return c * HDIM + (k ^ ((c & 7) << 3) ^ ((c & 8) >> 1));
}

// ---------------------------------------------------------------------------
// GEMM: out[nrows x 128] = f(A) [nrows x 128] @ W [128 x 128] + bias
// f(A) = APPLYT ? relu(A*sc + sh) (per input column) : A
// 256 threads = 8 waves; each wave computes a 16-row x 128-col strip with
// V_WMMA_F32_16X16X4_F32 (K stepped by 4, 8 N-tiles of 16).
// ---------------------------------------------------------------------------
template <bool APPLYT>
__global__ __launch_bounds__(256) void gemm128_wmma(
    const float* __restrict__ A, const float* __restrict__ W,
    const float* __restrict__ bias, float* __restrict__ out,
    const float* __restrict__ sc, const float* __restrict__ sh,
    int nrows)
{
  __shared__ float ldsW[HDIM * HDIM];     // 64 KB, swizzled-transposed [c][k]
  const int tid = threadIdx.x;
  for (int idx = tid; idx < HDIM * HDIM; idx += 256) {
    const int k = idx >> 7;
    const int n = idx & (HDIM - 1);
    ldsW[swzW(n, k)] = W[idx];
  }
  __syncthreads();

  const int wave = tid >> 5;
  const int lane = tid & 31;
  const int n16  = lane & 15;   // M for A-frag, N for B/C-frags
  const int kh   = lane >> 4;   // which K pair (0 -> K=0,1 ; 1 -> K=2,3)
  const int row0 = blockIdx.x * 128 + wave * 16;
  if (row0 >= nrows) return;    // nrows % 16 == 0 -> whole strips only

  const float* arow = A + (size_t)(row0 + n16) * HDIM;

  v8f zero = {};
  v8f acc[8];
#pragma unroll
  for (int t = 0; t < 8; ++t) acc[t] = zero;

  for (int k = 0; k < HDIM; k += 4) {
    const int kx = k + 2 * kh;
    // A-frag 16x4 f32: lane m=n16 holds (K=kx, K=kx+1)
    v2f a = *(const v2f*)(arow + kx);
    if (APPLYT) {
      a.x = fmaxf(a.x * sc[kx]     + sh[kx],     0.0f);
      a.y = fmaxf(a.y * sc[kx + 1] + sh[kx + 1], 0.0f);
    }
#pragma unroll
    for (int t = 0; t < 8; ++t) {
      const int c = t * 16 + n16;
      // B-frag 4x16 f32: lane col c holds (K=kx, K=kx+1) -> one ds_load_b64
      const v2f b = *(const v2f*)&ldsW[swzW(c, kx)];
      acc[t] = __builtin_amdgcn_wmma_f32_16x16x4_f32(
          false, a, false, b, (short)0, acc[t], false, false);
    }
  }

  // C/D 16x16 f32 layout: VGPR v -> row v (lanes 0-15) / row v+8 (lanes 16-31)
#pragma unroll
  for (int t = 0; t < 8; ++t) {
    const int c = t * 16 + n16;
    const float bz = bias[c];
#pragma unroll
    for (int v = 0; v < 8; ++v) {
      out[(size_t)(row0 + v + 8 * kh) * HDIM + c] = acc[t][v] + bz;
    }
  }
}

// ---------------------------------------------------------------------------
// z[dst] += h[src] over edges (z pre-initialized to h => z = h + agg).
// 32 consecutive lanes cover one edge row (32 float4 quads = 512 B).
// ---------------------------------------------------------------------------
__global__ void scatter_add_kernel(const float* __restrict__ h,
                                   const int* __restrict__ src,
                                   const int* __restrict__ dst,
                                   float* __restrict__ z, int nedges)
{
  const long total = (long)nedges * 32;       // quads of 4 floats
  const long stride = (long)gridDim.x * blockDim.x;
  for (long i = (long)blockIdx.x * blockDim.x + threadIdx.x; i < total; i += stride) {
    const int e = (int)(i >> 5);
    const int q = (int)(i & 31);
    // prefetch next grid-stride iteration's index words
    const long nx = i + stride;
    if (nx < total) {
      __builtin_prefetch(&src[nx >> 5], 0, 1);
      __builtin_prefetch(&dst[nx >> 5], 0, 1);
    }
    const int s = src[e];
    const int d = dst[e];
    const float4 v = *(const float4*)(h + (size_t)s * HDIM + q * 4);
    float* zp = z + (size_t)d * HDIM + q * 4;
    atomicAdd(zp + 0, v.x);
    atomicAdd(zp + 1, v.y);
    atomicAdd(zp + 2, v.z);
    atomicAdd(zp + 3, v.w);
  }
}

// ---------------------------------------------------------------------------
// Per-column sum and sum-of-squares (for batch-stat BN), float4 per thread.
// ---------------------------------------------------------------------------
__global__ void col_stats_kernel(const float* __restrict__ z,
                                 float* __restrict__ cs, float* __restrict__ cq,
                                 int nrows)
{
  const int g = blockIdx.x * blockDim.x + threadIdx.x;
  const int q = g & 31;                     // column quad (4 floats)
  const int r0 = g >> 5;
  const int rstride = (gridDim.x * blockDim.x) >> 5;
  float s0 = 0, s1 = 0, s2 = 0, s3 = 0;
  float q0 = 0, q1 = 0, q2 = 0, q3 = 0;
  for (int r = r0; r < nrows; r += rstride) {
    const float4 v = *(const float4*)(z + (size_t)r * HDIM + q * 4);
    s0 += v.x; q0 += v.x * v.x;
    s1 += v.y; q1 += v.y * v.y;
    s2 += v.z; q2 += v.z * v.z;
    s3 += v.w; q3 += v.w * v.w;
  }
  const int c = q * 4;
  atomicAdd(&cs[c + 0], s0); atomicAdd(&cq[c + 0], q0);
  atomicAdd(&cs[c + 1], s1); atomicAdd(&cq[c + 1], q1);
  atomicAdd(&cs[c + 2], s2); atomicAdd(&cq[c + 2], q2);
  atomicAdd(&cs[c + 3], s3); atomicAdd(&cq[c + 3], q3);
}

__global__ void bn_finalize_kernel(const float* __restrict__ cs,
                                   const float* __restrict__ cq,
                                   const float* __restrict__ gamma,
                                   const float* __restrict__ beta,
                                   float* __restrict__ sc, float* __restrict__ sh,
                                   float invN)
{
  const int c = threadIdx.x;           // 128 threads
  const float mu  = cs[c] * invN;
  const float var = cq[c] * invN - mu * mu;
  const float inv = rsqrtf(var + 1e-5f);
  const float s = gamma[c] * inv;
  sc[c] = s;
  sh[c] = beta[c] - mu * s;
}

__global__ void bn_relu_kernel(const float* __restrict__ z,
                               const float* __restrict__ sc,
                               const float* __restrict__ sh,
                               float* __restrict__ out, int totalQuads)
{
  const int stride = gridDim.x * blockDim.x;
  for (int i = blockIdx.x * blockDim.x + threadIdx.x; i < totalQuads; i += stride) {
    const int c = (i & 31) * 4;
    const float4 v = *(const float4*)(z + (size_t)i * 4);
    float4 r;
    r.x = fmaxf(v.x * sc[c + 0] + sh[c + 0], 0.0f);
    r.y = fmaxf(v.y * sc[c + 1] + sh[c + 1], 0.0f);
    r.z = fmaxf(v.z * sc[c + 2] + sh[c + 2], 0.0f);
    r.w = fmaxf(v.w * sc[c + 3] + sh[c + 3], 0.0f);
    *(float4*)(out + (size_t)i * 4) = r;
  }
}

__global__ void zero_f_kernel(float* __restrict__ p, int n)
{
  const int stride = gridDim.x * blockDim.x;
  for (int i = blockIdx.x * blockDim.x + threadIdx.x; i < n; i += stride)
    p[i] = 0.0f;
}

// ---------------------------------------------------------------------------
// Pooling: pooled[g][0:128]=sum h0, [128:256]=sum h1; counts per graph
// ---------------------------------------------------------------------------
__global__ void pool_accum_kernel(const float* __restrict__ h0,
                                  const float* __restrict__ h1,
                                  const int* __restrict__ batch,
                                  float* __restrict__ pooled, int nrows)
{
  const long total = (long)nrows * 64;        // 64 quads per node (2*128/4)
  const long stride = (long)gridDim.x * blockDim.x;
  for (long i = (long)blockIdx.x * blockDim.x + threadIdx.x; i < total; i += stride) {
    const int r = (int)(i >> 6);
    const int q = (int)(i & 63);
    const int g = batch[r];
    const float4 v = (q < 32)
        ? *(const float4*)(h0 + (size_t)r * HDIM + q * 4)
        : *(const float4*)(h1 + (size_t)r * HDIM + (q - 32) * 4);
    float* pp = pooled + (size_t)g * 256 + q * 4;
    atomicAdd(pp + 0, v.x);
    atomicAdd(pp + 1, v.y);
    atomicAdd(pp + 2, v.z);
    atomicAdd(pp + 3, v.w);
  }
}

__global__ void count_accum_kernel(const int* __restrict__ batch,
                                   float* __restrict__ counts, int nrows)
{
  const int stride = gridDim.x * blockDim.x;
  for (int r = blockIdx.x * blockDim.x + threadIdx.x; r < nrows; r += stride)
    atomicAdd(&counts[batch[r]], 1.0f);
}

__global__ __launch_bounds__(256) void pool_final_kernel(
    const float* __restrict__ pooled, const float* __restrict__ counts,
    const float* __restrict__ jkw, const float* __restrict__ jkb,
    float* __restrict__ out)
{
  __shared__ float red[256];
  const int g = blockIdx.x;
  const int c = threadIdx.x;
  const float cnt = fmaxf(counts[g], 1.0f);
  red[c] = pooled[(size_t)g * 256 + c] / cnt * jkw[c];
  __syncthreads();
  for (int s = 128; s > 0; s >>= 1) {
    if (c < s) red[c] += red[c + s];
    __syncthreads();
  }
  if (c == 0) out[g] = red[0] + jkb[0];
}

// ---------------------------------------------------------------------------
extern "C" void kernel_launch(void* const* d_in, const int* in_sizes, int n_in,
                              void* d_out, int out_size, void* d_ws, size_t ws_size,
                              hipStream_t stream)
{
  const float* x      = (const float*)d_in[0];
  const int*   edge   = (const int*)d_in[1];
  const int*   batch  = (const int*)d_in[2];
  const float* lin_w  = (const float*)d_in[3];
  const float* lin_b  = (const float*)d_in[4];
  const float* w1     = (const float*)d_in[5];
  const float* b1     = (const float*)d_in[6];
  const float* bn1g   = (const float*)d_in[7];
  const float* bn1b   = (const float*)d_in[8];
  const float* w2     = (const float*)d_in[9];
  const float* b2     = (const float*)d_in[10];
  const float* bng    = (const float*)d_in[11];
  const float* bnb    = (const float*)d_in[12];
  const float* jkw    = (const float*)d_in[13];
  const float* jkb    = (const float*)d_in[14];
  float* out = (float*)d_out;

  const int* src = edge;
  const int* dst = edge + NEDGES;

  const size_t NH = (size_t)NNODES * HDIM;
  float* ws = (float*)d_ws;
  float* h0     = ws;            ws += NH;
  float* zb     = ws;            ws += NH;
  float* t1     = ws;            ws += NH;
  float* t2     = ws;            ws += NH;
  float* ho0    = ws;            ws += NH;
  float* ho1    = ws;            ws += NH;
  float* cs     = ws;            ws += HDIM;   // cs,cq contiguous (zeroed together)
  float* cq     = ws;            ws += HDIM;
  float* scbuf  = ws;            ws += HDIM;
  float* shbuf  = ws;            ws += HDIM;
  float* pooled = ws;            ws += (size_t)NGRAPH * 256; // pooled,counts contiguous
  float* counts = ws;            ws += NGRAPH;

  const int gemmGrid = (NNODES + 127) / 128;  // 391 blocks, 8 waves x 16 rows each
  const float invN = 1.0f / (float)NNODES;

  // h0 = x @ lin_in_w + lin_in_b
  gemm128_wmma<false><<<gemmGrid, 256, 0, stream>>>(x, lin_w, lin_b, h0,
                                                    nullptr, nullptr, NNODES);

  const float* hin = h0;
  float* houts[LAYERS] = {ho0, ho1};
  for (int l = 0; l < LAYERS; ++l) {
    // z = h + segment_sum(h[src] -> dst)
    hipMemcpyAsync(zb, hin, NH * sizeof(float), hipMemcpyDeviceToDevice, stream);
    scatter_add_kernel<<<8192, 256, 0, stream>>>(hin, src, dst, zb, NEDGES);

    // t1 = z @ w1[l] + b1[l]
    gemm128_wmma<false><<<gemmGrid, 256, 0, stream>>>(
        zb, w1 + (size_t)l * HDIM * HDIM, b1 + l * HDIM, t1,
        nullptr, nullptr, NNODES);

    // inner BN stats -> scale/shift (ReLU fused into next GEMM's A-load)
    zero_f_kernel<<<1, 256, 0, stream>>>(cs, 2 * HDIM);
    col_stats_kernel<<<512, 256, 0, stream>>>(t1, cs, cq, NNODES);
    bn_finalize_kernel<<<1, HDIM, 0, stream>>>(cs, cq, bn1g + l * HDIM,
                                               bn1b + l * HDIM, scbuf, shbuf, invN);

    // t2 = relu(bn1(t1)) @ w2[l] + b2[l]   (transform fused into A-load)
    gemm128_wmma<true><<<gemmGrid, 256, 0, stream>>>(
        t1, w2 + (size_t)l * HDIM * HDIM, b2 + l * HDIM, t2,
        scbuf, shbuf, NNODES);

    // outer BN stats, then materialize h = relu(bn(t2)) (this is outs[l])
    zero_f_kernel<<<1, 256, 0, stream>>>(cs, 2 * HDIM);
    col_stats_kernel<<<512, 256, 0, stream>>>(t2, cs, cq, NNODES);
    bn_finalize_kernel<<<1, HDIM, 0, stream>>>(cs, cq, bng + l * HDIM,
                                               bnb + l * HDIM, scbuf, shbuf, invN);
    bn_relu_kernel<<<4096, 256, 0, stream>>>(t2, scbuf, shbuf, houts[l],
                                             (int)(NH / 4));
    hin = houts[l];
  }

  // global mean pool over [ho0 | ho1] then @ jk_w + jk_b
  zero_f_kernel<<<(NGRAPH * 256 + NGRAPH + 255) / 256, 256, 0, stream>>>(
      pooled, NGRAPH * 256 + NGRAPH);
  pool_accum_kernel<<<4096, 256, 0, stream>>>(ho0, ho1, batch, pooled, NNODES);
  count_accum_kernel<<<64, 256, 0, stream>>>(batch, counts, NNODES);
  pool_final_kernel<<<NGRAPH, 256, 0, stream>>>(pooled, counts, jkw, jkb, out);
}